// PhenoConnect7_74302934221044
// MI455X (gfx1250) — compile-verified
//
#include <hip/hip_runtime.h>
#include <hip/hip_bf16.h>
#include <math.h>

typedef __attribute__((ext_vector_type(2))) float v2f;
typedef __attribute__((ext_vector_type(8))) float v8f;
typedef __attribute__((ext_vector_type(4))) int   v4i;

#define G_LEN_F  20000.0f
#define H_LEN    16384
#define HIDDEN   32
#define BATCH    256
#define TK       128           // K-chunk staged in LDS per iteration
#define TSTRIDE  (TK + 4)      // pad: stride 132 -> bank stride 4, conflict-free fragment gathers
#define ROWS     16            // batch rows per block (one WMMA M-tile)
#define KSPLIT   16            // K-dimension split across blocks (for BW occupancy)
#define KRANGE   (H_LEN / KSPLIT)   // 1024 columns per block
#define KCHUNKS  (KRANGE / TK)      // 8 LDS chunks per block

// gfx1250 async global->LDS path (ASYNCcnt-tracked DMA into LDS, no VGPR
// round trip). Guarded: falls back to load+ds_store if the toolchain does
// not declare the builtins (host pass always takes the fallback).
// Probe feedback: param0 is a non-const 16B int-vector pointer in the global
// (addrspace(1), printed "__device__") address space; param1 is the LDS-side
// pointer (addrspace(3)); args 2/3 are imm offset / imm cpol.
#if defined(__AMDGCN__) && \
    __has_builtin(__builtin_amdgcn_global_load_async_to_lds_b128) && \
    __has_builtin(__builtin_amdgcn_s_wait_asynccnt)
#define USE_ASYNC 1
#define AS1 __attribute__((address_space(1)))
#define AS3 __attribute__((address_space(3)))
#define ASYNC_B128(gsrc_, ldst_) \
    __builtin_amdgcn_global_load_async_to_lds_b128( \
        (AS1 v4i*)(gsrc_), (AS3 v4i*)(ldst_), 0, 0)
#else
#define USE_ASYNC 0
#endif

// ---------------------------------------------------------------------------
// Stage 1: partial x3[b,c] for one (M-tile, K-split) via V_WMMA_F32_16X16X4_F32
// grid = (16 M-blocks, 16 K-splits), block = 256 threads (8 waves)
// ---------------------------------------------------------------------------
__global__ __launch_bounds__(256) void pheno_stage1(
    const float* __restrict__ gh1, const float* __restrict__ h1,
    const float* __restrict__ gh2, const float* __restrict__ h2,
    const float* __restrict__ hpo_par,
    const float* __restrict__ genes_hpo_scale,
    const float* __restrict__ hpo_bias,
    const float* __restrict__ gen_bias,
    float* __restrict__ ws_partial)   // [KSPLIT][BATCH][HIDDEN]
{
    // One shared pool: 4 input tiles (16 x TSTRIDE) + p tile (32 x TSTRIDE).
    // After the K loop the front of the pool is reused as the cross-wave
    // reduction buffer (4096 floats).  Total 96*132*4 = 50688 B < 64 KB.
    __shared__ float smem[(4 * ROWS + HIDDEN) * TSTRIDE];
    float* tg1 = smem + 0 * ROWS * TSTRIDE;
    float* th1 = smem + 1 * ROWS * TSTRIDE;
    float* tg2 = smem + 2 * ROWS * TSTRIDE;
    float* th2 = smem + 3 * ROWS * TSTRIDE;
    float* tp2 = smem + 4 * ROWS * TSTRIDE;
    float* red = smem;                       // aliased after final barrier

    const int tid   = threadIdx.x;
    const int lane  = tid & 31;
    const int wave  = tid >> 5;
    const int row0  = blockIdx.x * ROWS;     // batch-tile origin
    const int kbase = blockIdx.y * KRANGE;   // K-split origin

    // Per-lane channel coefficients for both 16-wide N tiles.
    // x1*x2 = a^2 g1g2 + ab(g1h2+h1g2) + ak(g1+g2) + b^2 h1h2 + bk(h1+h2) + k^2
    float c1[2], c2[2], c3[2], c4[2], c5[2], c6[2];
#pragma unroll
    for (int T = 0; T < 2; ++T) {
        const int ch = (lane & 15) + 16 * T;
        const float s  = genes_hpo_scale[ch];
        const float eg = expf(gen_bias[ch]);
        const float k  = G_LEN_F * s * expf(hpo_bias[ch]) * (1.0f + eg);
        const float a  = s;
        const float bb = s * eg;
        c1[T] = a * a;   c2[T] = a * bb;  c3[T] = a * k;
        c4[T] = bb * bb; c5[T] = bb * k;  c6[T] = k * k;
    }

    // Two accumulators per N-tile (odd/even terms) to break the WMMA C-chain.
    v8f accA[2] = {{}, {}};
    v8f accB[2] = {{}, {}};

    const int m    = lane & 15;       // A-fragment row  (M)
    const int koff = (lane >> 4) * 2; // K-pair selector (lanes 16-31 -> K=2,3)
    const v2f ones = {1.0f, 1.0f};

    for (int kc = 0; kc < KCHUNKS; ++kc) {
        const int k0 = kbase + kc * TK;
        __syncthreads();   // protect LDS reuse from previous chunk's compute

        const float* const gsrc[4] = {gh1, h1, gh2, h2};
        float* const       gdst[4] = {tg1, th1, tg2, th2};

#if USE_ASYNC
        // ---- DMA the five tiles straight into LDS (ASYNCcnt path) --------
#pragma unroll
        for (int aidx = 0; aidx < 4; ++aidx) {
#pragma unroll
            for (int i = 0; i < 2; ++i) {
                const int idx = tid + i * 256;       // 0..511 = 16 rows x 32 f4
                const int r   = idx >> 5;
                const int c4i = idx & 31;
                ASYNC_B128(gsrc[aidx] + (size_t)(row0 + r) * H_LEN + k0 + 4 * c4i,
                           gdst[aidx] + r * TSTRIDE + 4 * c4i);
            }
        }
#pragma unroll
        for (int i = 0; i < 4; ++i) {
            const int idx = tid + i * 256;           // 0..1023 = 32 rows x 32 f4
            const int r   = idx >> 5;
            const int c4i = idx & 31;
            ASYNC_B128(hpo_par + (size_t)r * H_LEN + k0 + 4 * c4i,
                       tp2 + r * TSTRIDE + 4 * c4i);
        }
        __builtin_amdgcn_s_wait_asynccnt(0);
#else
        // ---- fallback: coalesced float4 loads + ds stores -----------------
#pragma unroll
        for (int aidx = 0; aidx < 4; ++aidx) {
#pragma unroll
            for (int i = 0; i < 2; ++i) {
                const int idx = tid + i * 256;
                const int r   = idx >> 5;
                const int c4i = idx & 31;
                const float4 v = *(const float4*)(gsrc[aidx] +
                                  (size_t)(row0 + r) * H_LEN + k0 + 4 * c4i);
                *(float4*)(gdst[aidx] + r * TSTRIDE + 4 * c4i) = v;
            }
        }
#pragma unroll
        for (int i = 0; i < 4; ++i) {
            const int idx = tid + i * 256;
            const int r   = idx >> 5;
            const int c4i = idx & 31;
            const float4 v = *(const float4*)(hpo_par +
                              (size_t)r * H_LEN + k0 + 4 * c4i);
            *(float4*)(tp2 + r * TSTRIDE + 4 * c4i) = v;
        }
#endif
        __syncthreads();

        // ---- WMMA over this chunk: each wave takes K-steps wave+8*i -------
#pragma unroll
        for (int i = 0; i < (TK / 4) / 8; ++i) {     // 4 steps per wave
            const int kl = 4 * (wave + 8 * i) + koff;

            const v2f g1  = *(const v2f*)(tg1 + m * TSTRIDE + kl);
            const v2f hh1 = *(const v2f*)(th1 + m * TSTRIDE + kl);
            const v2f g2  = *(const v2f*)(tg2 + m * TSTRIDE + kl);
            const v2f hh2 = *(const v2f*)(th2 + m * TSTRIDE + kl);

            const v2f P1 = g1 * g2;
            const v2f P2 = g1 * hh2 + hh1 * g2;
            const v2f P3 = g1 + g2;
            const v2f P4 = hh1 * hh2;
            const v2f P5 = hh1 + hh2;

#pragma unroll
            for (int T = 0; T < 2; ++T) {
                const v2f p = *(const v2f*)(tp2 + ((lane & 15) + 16 * T) * TSTRIDE + kl);
                const v2f q = p * p;   // hpo_par^2 (DMA cannot transform)
                accA[T] = __builtin_amdgcn_wmma_f32_16x16x4_f32(
                    false, P1, false, q * c1[T], (short)0, accA[T], false, false);
                accB[T] = __builtin_amdgcn_wmma_f32_16x16x4_f32(
                    false, P2, false, q * c2[T], (short)0, accB[T], false, false);
                accA[T] = __builtin_amdgcn_wmma_f32_16x16x4_f32(
                    false, P3, false, q * c3[T], (short)0, accA[T], false, false);
                accB[T] = __builtin_amdgcn_wmma_f32_16x16x4_f32(
                    false, P4, false, q * c4[T], (short)0, accB[T], false, false);
                accA[T] = __builtin_amdgcn_wmma_f32_16x16x4_f32(
                    false, P5, false, q * c5[T], (short)0, accA[T], false, false);
                accB[T] = __builtin_amdgcn_wmma_f32_16x16x4_f32(
                    false, ones, false, q * c6[T], (short)0, accB[T], false, false);
            }
        }
    }

    // ---- cross-wave reduction of the 16x32 partial, deterministic output --
    __syncthreads();
#pragma unroll
    for (int T = 0; T < 2; ++T) {
        const v8f tot = accA[T] + accB[T];
#pragma unroll
        for (int j = 0; j < 8; ++j)
            red[wave * 512 + T * 256 + j * 32 + lane] = tot[j];
    }
    __syncthreads();
#pragma unroll
    for (int i = 0; i < 2; ++i) {
        const int o = tid + i * 256;                 // 0..511 outputs
        float x3 = 0.0f;
#pragma unroll
        for (int w = 0; w < 8; ++w) x3 += red[w * 512 + o];
        // C/D layout: VGPR j, lane l -> M = j + 8*(l>=16), N = l&15 (+16*tile)
        const int T  = o >> 8;
        const int j  = (o >> 5) & 7;
        const int l  = o & 31;
        const int bl = j + 8 * (l >> 4);
        const int ch = (l & 15) + 16 * T;
        ws_partial[(size_t)blockIdx.y * (BATCH * HIDDEN) +
                   (size_t)(row0 + bl) * HIDDEN + ch] = x3;
    }
}

// ---------------------------------------------------------------------------
// Stage 2: reduce K-split partials, sqrt/scale/bias, tiny MLP + sigmoid.
// 256 rows -> 1 block x 256 threads (one row per thread). Negligible work.
// ---------------------------------------------------------------------------
__global__ __launch_bounds__(256) void pheno_stage2(
    const float* __restrict__ ws_partial,
    const float* __restrict__ out_scale, const float* __restrict__ out_bias,
    const float* __restrict__ W1, const float* __restrict__ b1,
    const float* __restrict__ W2, const float* __restrict__ b2,
    float* __restrict__ y)
{
    __shared__ float sW1[64 * 32];
    __shared__ float sb1[64];
    __shared__ float sW2[64];
    const int tid = threadIdx.x;
    for (int i = tid; i < 64 * 32; i += 256) sW1[i] = W1[i];
    if (tid < 64) { sb1[tid] = b1[tid]; sW2[tid] = W2[tid]; }
    __syncthreads();

    const int b = tid;                 // batch row
    float x3[HIDDEN];
#pragma unroll
    for (int c = 0; c < HIDDEN; ++c) x3[c] = 0.0f;
    for (int kz = 0; kz < KSPLIT; ++kz) {
        const float* row = ws_partial + (size_t)kz * (BATCH * HIDDEN) +
                           (size_t)b * HIDDEN;
#pragma unroll
        for (int c4i = 0; c4i < HIDDEN / 4; ++c4i) {
            const float4 v = *(const float4*)(row + 4 * c4i);
            x3[4 * c4i + 0] += v.x; x3[4 * c4i + 1] += v.y;
            x3[4 * c4i + 2] += v.z; x3[4 * c4i + 3] += v.w;
        }
    }
    float x[HIDDEN];
#pragma unroll
    for (int c = 0; c < HIDDEN; ++c)
        x[c] = sqrtf(x3[c]) * out_scale[c] + out_bias[c];

    float acc = b2[0];
#pragma unroll 4
    for (int j = 0; j < 64; ++j) {
        float hj = sb1[j];
#pragma unroll
        for (int c = 0; c < HIDDEN; ++c) hj = fmaf(sW1[j * 32 + c], x[c], hj);
        hj = hj > 0.0f ? hj : 0.1f * hj;          // leaky-relu 0.1
        acc = fmaf(sW2[j], hj, acc);
    }
    y[b] = 1.0f / (1.0f + expf(-acc));            // sigmoid
}

// ---------------------------------------------------------------------------
extern "C" void kernel_launch(void* const* d_in, const int* in_sizes, int n_in,
                              void* d_out, int out_size, void* d_ws, size_t ws_size,
                              hipStream_t stream) {
    const float* gh1 = (const float*)d_in[0];
    const float* h1  = (const float*)d_in[1];
    const float* gh2 = (const float*)d_in[2];
    const float* h2  = (const float*)d_in[3];
    const float* hpo = (const float*)d_in[4];
    const float* osc = (const float*)d_in[5];
    const float* obi = (const float*)d_in[6];
    const float* ghs = (const float*)d_in[7];
    const float* hpb = (const float*)d_in[8];
    const float* gnb = (const float*)d_in[9];
    const float* W1  = (const float*)d_in[10];
    const float* b1  = (const float*)d_in[11];
    const float* W2  = (const float*)d_in[12];
    const float* b2  = (const float*)d_in[13];

    float* ws = (float*)d_ws;   // KSPLIT * 256 * 32 floats = 512 KB of partials

    dim3 g1(BATCH / ROWS, KSPLIT);   // 16 x 16 = 256 workgroups
    pheno_stage1<<<g1, 256, 0, stream>>>(gh1, h1, gh2, h2, hpo,
                                         ghs, hpb, gnb, ws);
    pheno_stage2<<<1, 256, 0, stream>>>(ws, osc, obi, W1, b1, W2, b2,
                                        (float*)d_out);
}